// MambaLayer_29592324670023
// MI455X (gfx1250) — compile-verified
//
#include <hip/hip_runtime.h>
#include <math.h>

// ---------------- problem constants ----------------
#define DIM      128
#define D_STATE  16
#define D_CONV   4
#define D_INNER  256          // 2*DIM
#define DT_RANK  8
#define NB       2            // batch
#define LSEQ     16384        // 16*32*32
#define LBITS    14           // LSEQ = 1<<14
#define BL       (NB*LSEQ)    // 32768
#define NPROJ    40           // DT_RANK + 2*D_STATE
#define NPROJ_PAD 48          // padded to 3 x 16 N-tiles

typedef float v2f __attribute__((ext_vector_type(2)));
typedef float v8f __attribute__((ext_vector_type(8)));

// ---------------------------------------------------------------
// Kernel 0: LayerNorm over channels.  x is (B, C, L); we produce
// xn as (BL, 128) row-major so GEMM1 A-loads are contiguous.
// Block = 128 threads handles a tile of 32 positions x 128 channels.
// ---------------------------------------------------------------
__global__ __launch_bounds__(128) void ln_kernel(const float* __restrict__ x,
                                                 const float* __restrict__ nw,
                                                 const float* __restrict__ nb,
                                                 float* __restrict__ xn) {
    __shared__ float tile[DIM][33];
    __shared__ float mu_s[32];
    __shared__ float rs_s[32];

    const int tid  = threadIdx.x;
    const int pos0 = blockIdx.x * 32;           // global position base
    const int b    = pos0 >> LBITS;
    const int l0   = pos0 & (LSEQ - 1);
    const size_t xbase = (size_t)b * DIM * LSEQ;

    // coalesced load of 128 x 32 tile of the (C, L) plane
    #pragma unroll 4
    for (int i = 0; i < 32; ++i) {
        int lin = i * 128 + tid;                // 0..4095
        int c   = lin >> 5;
        int col = lin & 31;
        tile[c][col] = x[xbase + (size_t)c * LSEQ + l0 + col];
    }
    __syncthreads();

    if (tid < 32) {
        float s = 0.f, s2 = 0.f;
        #pragma unroll 8
        for (int c = 0; c < DIM; ++c) {
            float v = tile[c][tid];
            s += v; s2 += v * v;
        }
        float mu  = s * (1.0f / DIM);
        float var = s2 * (1.0f / DIM) - mu * mu;
        mu_s[tid] = mu;
        rs_s[tid] = rsqrtf(var + 1e-6f);
    }
    __syncthreads();

    const float w  = nw[tid];
    const float bb = nb[tid];
    #pragma unroll 4
    for (int i = 0; i < 32; ++i) {              // row i = position pos0+i
        float v = (tile[tid][i] - mu_s[i]) * rs_s[i] * w + bb;
        xn[(size_t)(pos0 + i) * DIM + tid] = v;
    }
}

// ---------------------------------------------------------------
// Zero-pad x_proj_w (40,256) -> (48,256) so the GEMM2 inner loop
// needs no per-iteration select.
// ---------------------------------------------------------------
__global__ __launch_bounds__(256) void pad_w_kernel(const float* __restrict__ w,
                                                    float* __restrict__ wp) {
    int i = blockIdx.x * 256 + threadIdx.x;     // 0 .. 48*256-1
    int n = i >> 8;                              // row (ldW = 256)
    wp[i] = (n < NPROJ) ? w[i] : 0.f;
}

// ---------------------------------------------------------------
// f32 WMMA GEMM:  Y[m,n] = sum_k A[m,k] * W[n,k]
// One wave per (16 x NT*16) tile; NT accumulators reuse each A frag.
// 2-deep software pipeline: fragments for K-step s+2 are loaded while
// the WMMAs of step s issue, so s_wait only covers loads 2 steps old.
// A-frag layout: lane m=lane&15, VGPR pair holds K = 2*(lane>>4)+{0,1}.
// C/D: VGPR r -> M = r + 8*(lane>>4), N = lane&15.
// ---------------------------------------------------------------
template<int NT>
__device__ __forceinline__ void gemm_tile(const float* Ap, const float* const* Wp,
                                          int K, v8f* acc) {
    float2 aB[2];
    float2 bB[2][NT];
    const int nsteps = K >> 2;

    // prologue: fill both pipeline slots
    aB[0] = *reinterpret_cast<const float2*>(Ap);
    #pragma unroll
    for (int j = 0; j < NT; ++j) bB[0][j] = *reinterpret_cast<const float2*>(Wp[j]);
    aB[1] = *reinterpret_cast<const float2*>(Ap + 4);
    #pragma unroll
    for (int j = 0; j < NT; ++j) bB[1][j] = *reinterpret_cast<const float2*>(Wp[j] + 4);

    for (int s = 0; s < nsteps - 2; ++s) {
        const int cur = s & 1;
        v2f a; a.x = aB[cur].x; a.y = aB[cur].y;
        #pragma unroll
        for (int j = 0; j < NT; ++j) {
            v2f bf; bf.x = bB[cur][j].x; bf.y = bB[cur][j].y;
            acc[j] = __builtin_amdgcn_wmma_f32_16x16x4_f32(false, a, false, bf,
                                                           (short)0, acc[j], false, false);
        }
        // refill this slot with fragments for step s+2
        const int kn = (s + 2) * 4;
        aB[cur] = *reinterpret_cast<const float2*>(Ap + kn);
        #pragma unroll
        for (int j = 0; j < NT; ++j)
            bB[cur][j] = *reinterpret_cast<const float2*>(Wp[j] + kn);
    }
    // epilogue: drain the last two slots
    #pragma unroll
    for (int e = 0; e < 2; ++e) {
        const int cur = (nsteps - 2 + e) & 1;
        v2f a; a.x = aB[cur].x; a.y = aB[cur].y;
        #pragma unroll
        for (int j = 0; j < NT; ++j) {
            v2f bf; bf.x = bB[cur][j].x; bf.y = bB[cur][j].y;
            acc[j] = __builtin_amdgcn_wmma_f32_16x16x4_f32(false, a, false, bf,
                                                           (short)0, acc[j], false, false);
        }
    }
}

template<int NT>
__global__ __launch_bounds__(32) void gemm_wmma_f32(const float* __restrict__ A,
                                                    const float* __restrict__ W,
                                                    float* __restrict__ Y,
                                                    int K, int ldA, int ldW, int ldY,
                                                    int Nvalid) {
    const int lane  = threadIdx.x;
    const int half  = lane >> 4;
    const int l16   = lane & 15;
    const int m     = blockIdx.x * 16 + l16;
    const int nbase = blockIdx.y * NT * 16;

    const float* Ap = A + (size_t)m * ldA + 2 * half;
    const float* Wp[NT];
    #pragma unroll
    for (int j = 0; j < NT; ++j)
        Wp[j] = W + (size_t)(nbase + j * 16 + l16) * ldW + 2 * half;

    const v8f zero = {0.f, 0.f, 0.f, 0.f, 0.f, 0.f, 0.f, 0.f};
    v8f acc[NT];
    #pragma unroll
    for (int j = 0; j < NT; ++j) acc[j] = zero;

    gemm_tile<NT>(Ap, Wp, K, acc);

    #pragma unroll
    for (int j = 0; j < NT; ++j) {
        int n = nbase + j * 16 + l16;
        if (n < Nvalid) {
            #pragma unroll
            for (int r = 0; r < 8; ++r) {
                int mr = blockIdx.x * 16 + r + 8 * half;
                Y[(size_t)mr * ldY + n] = acc[j][r];
            }
        }
    }
}

// ---------------------------------------------------------------
// Kernel 3: depthwise causal conv (width 4) over L + SiLU.
// xs = xz[:, 0:256].  One block per position, thread = channel d.
// ---------------------------------------------------------------
__global__ __launch_bounds__(256) void conv_silu_kernel(const float* __restrict__ xz,
                                                        const float* __restrict__ cw,
                                                        const float* __restrict__ cb,
                                                        float* __restrict__ x_in) {
    const int pos = blockIdx.x;
    const int d   = threadIdx.x;
    const int b   = pos >> LBITS;
    const int l   = pos & (LSEQ - 1);
    float acc = cb[d];
    #pragma unroll
    for (int k = 0; k < D_CONV; ++k) {
        int lk = l - (D_CONV - 1) + k;
        if (lk >= 0)
            acc = fmaf(xz[((size_t)(b << LBITS) + lk) * 512 + d], cw[d * D_CONV + k], acc);
    }
    x_in[(size_t)pos * D_INNER + d] = acc / (1.0f + expf(-acc));
}

// ---------------------------------------------------------------
// Kernel 5: dt = softplus(dt_raw @ dt_proj_w.T + b), K = 8 (VALU).
// ---------------------------------------------------------------
__global__ __launch_bounds__(256) void dtproj_kernel(const float* __restrict__ x_dbl,
                                                     const float* __restrict__ dtw,
                                                     const float* __restrict__ dtb,
                                                     float* __restrict__ dt_full) {
    const int m = blockIdx.x;
    const int d = threadIdx.x;
    const float* row = x_dbl + (size_t)m * NPROJ;
    const float* wp  = dtw + d * DT_RANK;
    float v = dtb[d];
    #pragma unroll
    for (int r = 0; r < DT_RANK; ++r) v = fmaf(row[r], wp[r], v);
    float sp = (v > 20.f) ? v : log1pf(expf(v));
    dt_full[(size_t)m * D_INNER + d] = sp;
}

// ---------------------------------------------------------------
// Kernel 6: selective scan. Wave handles 2 (b,d) channels; 16 lanes
// per channel hold the 16 states. Fuses +x*D and the SiLU(z) gate.
// ---------------------------------------------------------------
__global__ __launch_bounds__(256) void scan_kernel(const float* __restrict__ xz,
                                                   const float* __restrict__ x_in,
                                                   const float* __restrict__ x_dbl,
                                                   const float* __restrict__ dt_full,
                                                   const float* __restrict__ A_log,
                                                   const float* __restrict__ Dp_,
                                                   float* __restrict__ yg) {
    const int tid  = threadIdx.x;
    const int lane = tid & 31;
    const int wave = tid >> 5;              // 0..7
    const int n    = lane & 15;             // state index
    const int dh   = lane >> 4;             // which of the 2 channels
    const int b    = blockIdx.y;
    const int d    = blockIdx.x * 16 + wave * 2 + dh;

    const float a  = -expf(A_log[d * D_STATE + n]);
    const float Dp = Dp_[d];
    float h = 0.f;
    const size_t base = (size_t)b * LSEQ;

    for (int t = 0; t < LSEQ; ++t) {
        const size_t row = base + t;
        float dtv = dt_full[row * D_INNER + d];
        float xv  = x_in  [row * D_INNER + d];
        float Bv  = x_dbl [row * NPROJ + DT_RANK + n];
        float Cv  = x_dbl [row * NPROJ + DT_RANK + D_STATE + n];

        if (t + 32 < LSEQ) {                 // stream-ahead hints
            __builtin_prefetch(&dt_full[(row + 32) * D_INNER + d], 0, 1);
            __builtin_prefetch(&x_dbl [(row + 32) * NPROJ + n],    0, 1);
        }

        float dA = expf(dtv * a);
        h = fmaf(dA, h, dtv * xv * Bv);
        float p = h * Cv;
        p += __shfl_xor(p, 1);
        p += __shfl_xor(p, 2);
        p += __shfl_xor(p, 4);
        p += __shfl_xor(p, 8);               // sum over 16-lane half
        if (n == 0) {
            float zv = xz[row * 512 + D_INNER + d];
            float sg = zv / (1.0f + expf(-zv));
            yg[row * D_INNER + d] = (p + xv * Dp) * sg;
        }
    }
}

// ---------------------------------------------------------------
// Kernel 7: out-proj WMMA GEMM (NT=4 N-tiles per wave, same 2-deep
// pipeline) with fused transpose store to (B,C,L).
// ---------------------------------------------------------------
__global__ __launch_bounds__(32) void gemm_out_wmma(const float* __restrict__ A,   // yg (BL,256)
                                                    const float* __restrict__ W,   // out_proj_w (128,256)
                                                    float* __restrict__ out) {     // (B,128,L)
    const int NT = 4;
    const int lane  = threadIdx.x;
    const int half  = lane >> 4;
    const int l16   = lane & 15;
    const int m     = blockIdx.x * 16 + l16;
    const int nbase = blockIdx.y * NT * 16;

    const float* Ap = A + (size_t)m * D_INNER + 2 * half;
    const float* Wp[NT];
    #pragma unroll
    for (int j = 0; j < NT; ++j)
        Wp[j] = W + (size_t)(nbase + j * 16 + l16) * D_INNER + 2 * half;

    const v8f zero = {0.f, 0.f, 0.f, 0.f, 0.f, 0.f, 0.f, 0.f};
    v8f acc[NT];
    #pragma unroll
    for (int j = 0; j < NT; ++j) acc[j] = zero;

    gemm_tile<NT>(Ap, Wp, D_INNER, acc);

    #pragma unroll
    for (int j = 0; j < NT; ++j) {
        int c = nbase + j * 16 + l16;
        #pragma unroll
        for (int r = 0; r < 8; ++r) {
            int pos = blockIdx.x * 16 + r + 8 * half;
            int bb  = pos >> LBITS;
            int l   = pos & (LSEQ - 1);
            out[((size_t)bb * DIM + c) * LSEQ + l] = acc[j][r];
        }
    }
}

// ---------------- workspace layout (floats) ----------------
#define XN_OFF    ((size_t)0)                          // (BL,128)
#define XZ_OFF    (XN_OFF  + (size_t)BL * DIM)         // (BL,512)
#define XIN_OFF   (XZ_OFF  + (size_t)BL * 512)         // (BL,256)
#define XDBL_OFF  (XIN_OFF + (size_t)BL * D_INNER)     // (BL,40)
#define DT_OFF    (XDBL_OFF+ (size_t)BL * NPROJ)       // (BL,256)
#define YG_OFF    (DT_OFF  + (size_t)BL * D_INNER)     // (BL,256)
#define WPAD_OFF  (YG_OFF  + (size_t)BL * D_INNER)     // (48,256)

extern "C" void kernel_launch(void* const* d_in, const int* in_sizes, int n_in,
                              void* d_out, int out_size, void* d_ws, size_t ws_size,
                              hipStream_t stream) {
    (void)in_sizes; (void)n_in; (void)out_size; (void)ws_size;
    const float* x         = (const float*)d_in[0];
    const float* norm_w    = (const float*)d_in[1];
    const float* norm_b    = (const float*)d_in[2];
    const float* in_proj_w = (const float*)d_in[3];
    const float* conv_w    = (const float*)d_in[4];
    const float* conv_b    = (const float*)d_in[5];
    const float* x_proj_w  = (const float*)d_in[6];
    const float* dt_proj_w = (const float*)d_in[7];
    const float* dt_proj_b = (const float*)d_in[8];
    const float* A_log     = (const float*)d_in[9];
    const float* D_param   = (const float*)d_in[10];
    const float* out_proj_w= (const float*)d_in[11];
    float* out = (float*)d_out;

    float* ws     = (float*)d_ws;
    float* xn     = ws + XN_OFF;
    float* xz     = ws + XZ_OFF;
    float* x_in   = ws + XIN_OFF;
    float* x_dbl  = ws + XDBL_OFF;
    float* dtf    = ws + DT_OFF;
    float* yg     = ws + YG_OFF;
    float* wpad   = ws + WPAD_OFF;

    // 0) zero-pad x_proj_w (40,256) -> (48,256)
    pad_w_kernel<<<NPROJ_PAD, 256, 0, stream>>>(x_proj_w, wpad);

    // 1) LayerNorm -> xn (BL,128)
    ln_kernel<<<BL / 32, 128, 0, stream>>>(x, norm_w, norm_b, xn);

    // 2) xz = xn @ in_proj_w.T   (BL,128)x(512,128)^T -> (BL,512)
    gemm_wmma_f32<4><<<dim3(BL / 16, 512 / (4 * 16)), 32, 0, stream>>>(
        xn, in_proj_w, xz, DIM, DIM, DIM, 512, 512);

    // 3) conv + silu -> x_in (BL,256)
    conv_silu_kernel<<<BL, D_INNER, 0, stream>>>(xz, conv_w, conv_b, x_in);

    // 4) x_dbl = x_in @ wpad.T   (BL,256)x(48,256)^T -> (BL,40 valid)
    gemm_wmma_f32<3><<<dim3(BL / 16, 1), 32, 0, stream>>>(
        x_in, wpad, x_dbl, D_INNER, D_INNER, D_INNER, NPROJ, NPROJ);

    // 5) dt = softplus(dt_raw @ dt_proj_w.T + b) -> (BL,256)
    dtproj_kernel<<<BL, D_INNER, 0, stream>>>(x_dbl, dt_proj_w, dt_proj_b, dtf);

    // 6) selective scan, fused +x*D and silu(z) gate -> yg (BL,256)
    scan_kernel<<<dim3(D_INNER / 16, NB), 256, 0, stream>>>(
        xz, x_in, x_dbl, dtf, A_log, D_param, yg);

    // 7) out = yg @ out_proj_w.T, stored as (B,C,L)
    gemm_out_wmma<<<dim3(BL / 16, DIM / (4 * 16)), 32, 0, stream>>>(yg, out_proj_w, out);
}